// Down_channelV2_86930138071596
// MI455X (gfx1250) — compile-verified
//
#include <hip/hip_runtime.h>
#include <hip/hip_bf16.h>

typedef __attribute__((ext_vector_type(2))) float v2f;
typedef __attribute__((ext_vector_type(8))) float v8f;

#define CC   288
#define DD   576
#define NN   16
#define RR   18
#define LL   4096
#define BB   2
#define OUTC 96
#define MTOT 8192            // B*L rows
#define LC   1179648         // L*C

// ---------------------------------------------------------------------------
// WMMA helper: D = A(16x4 f32) * B(4x16 f32) + C, accumulate fp32
// ---------------------------------------------------------------------------
__device__ __forceinline__ v8f wmma4(v8f acc, v2f a, v2f b) {
    return __builtin_amdgcn_wmma_f32_16x16x4_f32(false, a, false, b,
                                                 (short)0, acc, false, false);
}

// ---------------------------------------------------------------------------
// Kernel 1: concat(x1,x2,x3) -> layernorm(C=288) -> xn  (row-major M x 288)
// ---------------------------------------------------------------------------
__global__ void ln0_kernel(const float* __restrict__ x1, const float* __restrict__ x2,
                           const float* __restrict__ x3, const float* __restrict__ g,
                           const float* __restrict__ bt, float* __restrict__ xn) {
    int row = blockIdx.x;            // b*4096 + l
    int c   = threadIdx.x;           // 0..287
    int b = row >> 12, l = row & 4095;
    const float* src = (c < 96) ? x1 : (c < 192 ? x2 : x3);
    int cm = c - (c < 96 ? 0 : (c < 192 ? 96 : 192));
    float v = src[(size_t)(b * 96 + cm) * LL + l];

    float s = v, s2 = v * v;
#pragma unroll
    for (int off = 16; off > 0; off >>= 1) {
        s  += __shfl_xor(s, off);
        s2 += __shfl_xor(s2, off);
    }
    __shared__ float ps[9], ps2[9], mu_s, var_s;
    int wid = c >> 5, lane = c & 31;
    if (lane == 0) { ps[wid] = s; ps2[wid] = s2; }
    __syncthreads();
    if (c == 0) {
        float ts = 0.f, ts2 = 0.f;
        for (int i = 0; i < 9; ++i) { ts += ps[i]; ts2 += ps2[i]; }
        float mu = ts * (1.f / 288.f);
        mu_s = mu;  var_s = ts2 * (1.f / 288.f) - mu * mu;
    }
    __syncthreads();
    float xnv = (v - mu_s) * rsqrtf(var_s + 1e-5f) * g[c] + bt[c];
    xn[(size_t)row * CC + c] = xnv;
}

// ---------------------------------------------------------------------------
// Kernel 2: xz = xn @ W_in   (M=8192, K=288, N=1152)  -- WMMA f32
// Each wave computes a 16x64 strip (4 accumulators) -> A fragment reused 4x.
// ---------------------------------------------------------------------------
__global__ void gemm_in_kernel(const float* __restrict__ A, const float* __restrict__ Bm,
                               float* __restrict__ Cm) {
    const int K = CC, N = 2 * DD;          // 288, 1152
    int wave = blockIdx.x * 8 + (threadIdx.x >> 5);
    int lane = threadIdx.x & 31;
    int ng = wave % (N / 64), tm = wave / (N / 64);   // 18 n-groups of 64
    int ml = lane & 15, hi = lane >> 4;
    const float* Ar = A + (size_t)(tm * 16 + ml) * K;
    int col0 = ng * 64 + ml;
    v8f acc[4];
#pragma unroll
    for (int j = 0; j < 4; ++j) acc[j] = (v8f){0.f,0.f,0.f,0.f,0.f,0.f,0.f,0.f};
    for (int k0 = 0; k0 < K; k0 += 4) {
        int k = k0 + 2 * hi;
        v2f a = *(const v2f*)(Ar + k);
        const float* B0 = Bm + (size_t)k * N + col0;
        const float* B1 = B0 + N;
#pragma unroll
        for (int j = 0; j < 4; ++j) {
            v2f b; b.x = B0[16 * j]; b.y = B1[16 * j];
            acc[j] = wmma4(acc[j], a, b);
        }
    }
#pragma unroll
    for (int j = 0; j < 4; ++j)
#pragma unroll
        for (int r = 0; r < 8; ++r) {
            int m = tm * 16 + r + 8 * hi;
            Cm[(size_t)m * N + col0 + 16 * j] = acc[j][r];
        }
}

// ---------------------------------------------------------------------------
// Kernel 3: depthwise causal conv (K=3) + bias + SiLU -> xc (M x 576)
// ---------------------------------------------------------------------------
__global__ void conv_silu_kernel(const float* __restrict__ xz, const float* __restrict__ cw,
                                 const float* __restrict__ cb, float* __restrict__ xc) {
    size_t idx = (size_t)blockIdx.x * blockDim.x + threadIdx.x;  // b*L*D + l*D + d
    int d = (int)(idx % DD);
    size_t row = idx / DD;            // b*L + l
    int l = (int)(row & 4095);
    float v = cb[d];
#pragma unroll
    for (int k = 0; k < 3; ++k) {
        int ll = l - 2 + k;
        if (ll >= 0) v += xz[(row - (size_t)(l - ll)) * (2 * DD) + d] * cw[d * 3 + k];
    }
    float s = v / (1.f + __expf(-v));
    xc[row * DD + d] = s;
}

// ---------------------------------------------------------------------------
// Kernel 4: x_dbl = xc @ W_x  (M=8192, K=576, realN=50 padded to 64) -- WMMA
// One wave covers all 64 padded cols (4 accumulators).
// ---------------------------------------------------------------------------
__global__ void gemm_wx_kernel(const float* __restrict__ A, const float* __restrict__ Wx,
                               float* __restrict__ Cm) {
    const int K = DD, Nr = RR + 2 * NN;   // 576, 50
    int wave = blockIdx.x * 8 + (threadIdx.x >> 5);
    int lane = threadIdx.x & 31;
    int tm = wave;                         // 512 m-tiles, 1 n-group
    int ml = lane & 15, hi = lane >> 4;
    const float* Ar = A + (size_t)(tm * 16 + ml) * K;
    v8f acc[4];
#pragma unroll
    for (int j = 0; j < 4; ++j) acc[j] = (v8f){0.f,0.f,0.f,0.f,0.f,0.f,0.f,0.f};
    for (int k0 = 0; k0 < K; k0 += 4) {
        int k = k0 + 2 * hi;
        v2f a = *(const v2f*)(Ar + k);
#pragma unroll
        for (int j = 0; j < 4; ++j) {
            int col = ml + 16 * j;
            v2f b;
            b.x = (col < Nr) ? Wx[(size_t)k * Nr + col] : 0.f;
            b.y = (col < Nr) ? Wx[(size_t)(k + 1) * Nr + col] : 0.f;
            acc[j] = wmma4(acc[j], a, b);
        }
    }
#pragma unroll
    for (int j = 0; j < 4; ++j) {
        int col = ml + 16 * j;
        if (col < Nr)
#pragma unroll
            for (int r = 0; r < 8; ++r) {
                int m = tm * 16 + r + 8 * hi;
                Cm[(size_t)m * Nr + col] = acc[j][r];
            }
    }
}

// ---------------------------------------------------------------------------
// Kernel 5: dt = softplus(x_dbl[:, :18] @ W_dt + b_dt) (K=18 pad 20, N=576)
// 16x64 strip per wave (4 accumulators).
// ---------------------------------------------------------------------------
__global__ void gemm_dt_kernel(const float* __restrict__ xdbl, const float* __restrict__ Wdt,
                               const float* __restrict__ bdt, float* __restrict__ dt) {
    const int N = DD, Nr = RR + 2 * NN;   // A ld = 50
    int wave = blockIdx.x * 8 + (threadIdx.x >> 5);
    int lane = threadIdx.x & 31;
    int ng = wave % (N / 64), tm = wave / (N / 64);   // 9 n-groups
    int ml = lane & 15, hi = lane >> 4;
    const float* Ar = xdbl + (size_t)(tm * 16 + ml) * Nr;
    int col0 = ng * 64 + ml;
    v8f acc[4];
#pragma unroll
    for (int j = 0; j < 4; ++j) acc[j] = (v8f){0.f,0.f,0.f,0.f,0.f,0.f,0.f,0.f};
    for (int k0 = 0; k0 < 20; k0 += 4) {
        int k = k0 + 2 * hi;
        v2f a;
        a.x = (k     < RR) ? Ar[k]     : 0.f;
        a.y = (k + 1 < RR) ? Ar[k + 1] : 0.f;
#pragma unroll
        for (int j = 0; j < 4; ++j) {
            int col = col0 + 16 * j;
            v2f b;
            b.x = (k     < RR) ? Wdt[(size_t)k * N + col]       : 0.f;
            b.y = (k + 1 < RR) ? Wdt[(size_t)(k + 1) * N + col] : 0.f;
            acc[j] = wmma4(acc[j], a, b);
        }
    }
#pragma unroll
    for (int j = 0; j < 4; ++j) {
        int col = col0 + 16 * j;
        float bias = bdt[col];
#pragma unroll
        for (int r = 0; r < 8; ++r) {
            int m = tm * 16 + r + 8 * hi;
            float v = acc[j][r] + bias;
            float sp = (v > 20.f) ? v : log1pf(__expf(v));
            dt[(size_t)m * N + col] = sp;
        }
    }
}

// ---------------------------------------------------------------------------
// Kernel 6: sequential selective scan over L, one thread per (b,d)
// ---------------------------------------------------------------------------
__global__ void scan_kernel(const float* __restrict__ dt, const float* __restrict__ xc,
                            const float* __restrict__ xz, const float* __restrict__ xdbl,
                            const float* __restrict__ A_log, const float* __restrict__ Dssm,
                            float* __restrict__ yg) {
    int tid = blockIdx.x * blockDim.x + threadIdx.x;   // 0..1151 exactly
    int b = tid / DD, d = tid % DD;
    float negA[NN], h[NN];
#pragma unroll
    for (int n = 0; n < NN; ++n) { negA[n] = -__expf(A_log[d * NN + n]); h[n] = 0.f; }
    float Dd = Dssm[d];
    size_t base576  = (size_t)b * LL * DD + d;
    size_t base1152 = (size_t)b * LL * (2 * DD) + DD + d;
    size_t base50   = (size_t)b * LL * 50;
    for (int t = 0; t < LL; ++t) {
        float dtv = dt[base576 + (size_t)t * DD];
        float xv  = xc[base576 + (size_t)t * DD];
        float zv  = xz[base1152 + (size_t)t * (2 * DD)];
        const float2* p = (const float2*)(xdbl + base50 + (size_t)t * 50 + RR);
        float bc[32];
#pragma unroll
        for (int j = 0; j < 16; ++j) { float2 q = p[j]; bc[2 * j] = q.x; bc[2 * j + 1] = q.y; }
        float du = dtv * xv;
        float y = 0.f;
#pragma unroll
        for (int n = 0; n < NN; ++n) {
            float dA = __expf(dtv * negA[n]);
            h[n] = dA * h[n] + du * bc[n];
            y += h[n] * bc[NN + n];
        }
        float sz = zv / (1.f + __expf(-zv));
        yg[base576 + (size_t)t * DD] = (y + xv * Dd) * sz;
    }
}

// ---------------------------------------------------------------------------
// Kernel 7: mamba_out = x_concat + y_gate @ W_out  (K=576, N=288) -- WMMA
// 16x48 strip per wave (3 accumulators).
// ---------------------------------------------------------------------------
__global__ void gemm_out_kernel(const float* __restrict__ A, const float* __restrict__ Wo,
                                const float* __restrict__ x1, const float* __restrict__ x2,
                                const float* __restrict__ x3, float* __restrict__ Cm) {
    const int K = DD, N = CC;
    int wave = blockIdx.x * 8 + (threadIdx.x >> 5);
    int lane = threadIdx.x & 31;
    int ng = wave % (N / 48), tm = wave / (N / 48);   // 6 n-groups of 48
    int ml = lane & 15, hi = lane >> 4;
    const float* Ar = A + (size_t)(tm * 16 + ml) * K;
    int col0 = ng * 48 + ml;
    v8f acc[3];
#pragma unroll
    for (int j = 0; j < 3; ++j) acc[j] = (v8f){0.f,0.f,0.f,0.f,0.f,0.f,0.f,0.f};
    for (int k0 = 0; k0 < K; k0 += 4) {
        int k = k0 + 2 * hi;
        v2f a = *(const v2f*)(Ar + k);
        const float* B0 = Wo + (size_t)k * N + col0;
        const float* B1 = B0 + N;
#pragma unroll
        for (int j = 0; j < 3; ++j) {
            v2f b; b.x = B0[16 * j]; b.y = B1[16 * j];
            acc[j] = wmma4(acc[j], a, b);
        }
    }
#pragma unroll
    for (int j = 0; j < 3; ++j) {
        int col = col0 + 16 * j;
        const float* src = (col < 96) ? x1 : (col < 192 ? x2 : x3);
        int cm = col - (col < 96 ? 0 : (col < 192 ? 96 : 192));
#pragma unroll
        for (int r = 0; r < 8; ++r) {
            int m = tm * 16 + r + 8 * hi;
            int bb = m >> 12, l = m & 4095;
            float res = src[(size_t)(bb * 96 + cm) * LL + l];
            Cm[(size_t)m * N + col] = acc[j][r] + res;
        }
    }
}

// ---------------------------------------------------------------------------
// Kernel 8: fc = reinterp(mamba) @ fc1_w + fc1_b  (K=288, N=96) -- WMMA
//   A[m=(b,hw)][k] = mamba_flat[b*L*C + k*4096 + hw]  (reshape+transpose fused)
// 16x48 strip per wave (3 accumulators), so the gathered A is reused 3x.
// ---------------------------------------------------------------------------
__global__ void gemm_fc1_kernel(const float* __restrict__ mamba, const float* __restrict__ W,
                                const float* __restrict__ bias, float* __restrict__ Cm) {
    const int K = CC, N = OUTC;
    int wave = blockIdx.x * 8 + (threadIdx.x >> 5);
    int lane = threadIdx.x & 31;
    int ng = wave % 2, tm = wave / 2;       // 2 n-groups of 48
    int ml = lane & 15, hi = lane >> 4;
    int bb = tm >> 8;                       // 256 m-tiles per batch
    int hw = ((tm & 255) << 4) + ml;
    size_t abase = (size_t)bb * LC + hw;
    int col0 = ng * 48 + ml;
    v8f acc[3];
#pragma unroll
    for (int j = 0; j < 3; ++j) acc[j] = (v8f){0.f,0.f,0.f,0.f,0.f,0.f,0.f,0.f};
    for (int k0 = 0; k0 < K; k0 += 4) {
        int k = k0 + 2 * hi;
        v2f a; a.x = mamba[abase + (size_t)k * LL]; a.y = mamba[abase + (size_t)(k + 1) * LL];
        const float* B0 = W + (size_t)k * N + col0;
        const float* B1 = B0 + N;
#pragma unroll
        for (int j = 0; j < 3; ++j) {
            v2f b; b.x = B0[16 * j]; b.y = B1[16 * j];
            acc[j] = wmma4(acc[j], a, b);
        }
    }
#pragma unroll
    for (int j = 0; j < 3; ++j) {
        int col = col0 + 16 * j;
        float bv = bias[col];
#pragma unroll
        for (int r = 0; r < 8; ++r) {
            int m = tm * 16 + r + 8 * hi;
            Cm[(size_t)m * N + col] = acc[j][r] + bv;
        }
    }
}

// ---------------------------------------------------------------------------
// Kernel 9: layernorm(96) + exact GELU + transpose to (B,96,H,W)
// ---------------------------------------------------------------------------
__global__ void ln1_gelu_kernel(const float* __restrict__ fc, const float* __restrict__ g,
                                const float* __restrict__ bt, float* __restrict__ out) {
    int row = blockIdx.x;           // b*4096 + hw
    int o   = threadIdx.x;          // 0..95
    float v = fc[(size_t)row * OUTC + o];
    float s = v, s2 = v * v;
#pragma unroll
    for (int off = 16; off > 0; off >>= 1) {
        s  += __shfl_xor(s, off);
        s2 += __shfl_xor(s2, off);
    }
    __shared__ float ps[3], ps2[3], mu_s, var_s;
    int wid = o >> 5, lane = o & 31;
    if (lane == 0) { ps[wid] = s; ps2[wid] = s2; }
    __syncthreads();
    if (o == 0) {
        float ts = ps[0] + ps[1] + ps[2];
        float ts2 = ps2[0] + ps2[1] + ps2[2];
        float mu = ts * (1.f / 96.f);
        mu_s = mu;  var_s = ts2 * (1.f / 96.f) - mu * mu;
    }
    __syncthreads();
    float x = (v - mu_s) * rsqrtf(var_s + 1e-5f) * g[o] + bt[o];
    float ge = 0.5f * x * (1.f + erff(x * 0.70710678118654752f));
    int b = row >> 12, hw = row & 4095;
    out[(size_t)b * (OUTC * LL) + (size_t)o * LL + hw] = ge;
}

// ---------------------------------------------------------------------------
extern "C" void kernel_launch(void* const* d_in, const int* in_sizes, int n_in,
                              void* d_out, int out_size, void* d_ws, size_t ws_size,
                              hipStream_t stream) {
    const float* x1    = (const float*)d_in[0];
    const float* x2    = (const float*)d_in[1];
    const float* x3    = (const float*)d_in[2];
    const float* ln0_g = (const float*)d_in[3];
    const float* ln0_b = (const float*)d_in[4];
    const float* W_in  = (const float*)d_in[5];
    const float* cw    = (const float*)d_in[6];
    const float* cb    = (const float*)d_in[7];
    const float* W_x   = (const float*)d_in[8];
    const float* W_dt  = (const float*)d_in[9];
    const float* b_dt  = (const float*)d_in[10];
    const float* A_log = (const float*)d_in[11];
    const float* D_ssm = (const float*)d_in[12];
    const float* W_out = (const float*)d_in[13];
    const float* fc1_w = (const float*)d_in[14];
    const float* fc1_b = (const float*)d_in[15];
    const float* ln1_g = (const float*)d_in[16];
    const float* ln1_b = (const float*)d_in[17];
    float* out = (float*)d_out;

    float* ws   = (float*)d_ws;
    float* xn   = ws;                                   //  8192*288
    float* xz   = xn   + (size_t)MTOT * CC;             //  8192*1152
    float* xc   = xz   + (size_t)MTOT * 2 * DD;         //  8192*576
    float* xdbl = xc   + (size_t)MTOT * DD;             //  8192*50
    float* dt   = xdbl + (size_t)MTOT * 50;             //  8192*576
    float* yg   = dt   + (size_t)MTOT * DD;             //  8192*576
    float* mamba = xn;                                  // reuse (xn dead after GEMM-in)
    float* fcout = xz;                                  // reuse (xz dead after scan)

    // 1. concat + LN(288)
    ln0_kernel<<<MTOT, 288, 0, stream>>>(x1, x2, x3, ln0_g, ln0_b, xn);
    // 2. xz = xn @ W_in          (512 m-tiles * 18 n-groups = 9216 waves)
    gemm_in_kernel<<<1152, 256, 0, stream>>>(xn, W_in, xz);
    // 3. depthwise conv + SiLU
    conv_silu_kernel<<<(MTOT * DD) / 256, 256, 0, stream>>>(xz, cw, cb, xc);
    // 4. x_dbl = xc @ W_x        (512 waves)
    gemm_wx_kernel<<<64, 256, 0, stream>>>(xc, W_x, xdbl);
    // 5. dt = softplus(dt_r @ W_dt + b_dt)   (512*9 = 4608 waves)
    gemm_dt_kernel<<<576, 256, 0, stream>>>(xdbl, W_dt, b_dt, dt);
    // 6. selective scan + gate   (1152 threads)
    scan_kernel<<<9, 128, 0, stream>>>(dt, xc, xz, xdbl, A_log, D_ssm, yg);
    // 7. mamba = x + yg @ W_out  (512*6 = 3072 waves)
    gemm_out_kernel<<<384, 256, 0, stream>>>(yg, W_out, x1, x2, x3, mamba);
    // 8. fc = reinterp(mamba) @ fc1_w + b    (512*2 = 1024 waves)
    gemm_fc1_kernel<<<128, 256, 0, stream>>>(mamba, fc1_w, fc1_b, fcout);
    // 9. LN(96) + GELU + transpose
    ln1_gelu_kernel<<<MTOT, 96, 0, stream>>>(fcout, ln1_g, ln1_b, out);
}